// TopologyTracker_20547123544587
// MI455X (gfx1250) — compile-verified
//
#include <hip/hip_runtime.h>

#define NUM_TILES 64
#define BINS (NUM_TILES * NUM_TILES)
#define BLOCK 256
#define CHUNK 2048            // events per TDM stage (per array)
#define NBLOCKS 512
#define SLICES 8              // reduce kernel: partial-histogram slices

typedef unsigned int u32x4 __attribute__((ext_vector_type(4)));
typedef int          i32x8 __attribute__((ext_vector_type(8)));
typedef int          i32x4 __attribute__((ext_vector_type(4)));

__device__ __forceinline__ unsigned lds_byte_offset(const void* p) {
    // addrspace(3) -> generic keeps the LDS byte offset in the low 32 bits
    return (unsigned)(unsigned long long)p;
}

// 1-D TDM descriptor: DMA `nelems` dwords from `gaddr` into LDS at `lds_addr`.
// D# group0: count=1, lds_addr, global_addr[56:0], type=2 ("image").
// D# group1: data_size=4B (code 2), tensor_dim0=tile_dim0=nelems, tensor_dim1=1.
__device__ __forceinline__ void tdm_load_1d(unsigned lds_addr, const void* gaddr,
                                            unsigned nelems) {
    unsigned long long ga = (unsigned long long)gaddr;
    u32x4 g0;
    g0.x = 1u;                                              // count=1 (valid user D#)
    g0.y = lds_addr;                                        // lds_addr (bytes)
    g0.z = (unsigned)(ga & 0xFFFFFFFFull);                  // global_addr[31:0]
    g0.w = (unsigned)((ga >> 32) & 0x01FFFFFFull)           // global_addr[56:32]
         | (2u << 30);                                      // type=2
    i32x8 g1;
    g1[0] = (int)(2u << 16);                                // data_size=2 -> 4 bytes
    g1[1] = (int)((nelems & 0xFFFFu) << 16);                // tensor_dim0[15:0] @bit48
    g1[2] = (int)(((nelems >> 16) & 0xFFFFu)                // tensor_dim0[31:16]
         | (1u << 16));                                     // tensor_dim1 = 1 @bit80
    g1[3] = (int)((nelems & 0xFFFFu) << 16);                // tile_dim0 @bit112
    g1[4] = 0;                                              // tile_dim1/2 unused
    g1[5] = (int)nelems;                                    // tensor_dim0_stride (unused for 1D)
    g1[6] = 0;
    g1[7] = 0;
    i32x4 z4 = {0, 0, 0, 0};                                // groups 2/3 unused (<=2D)
    i32x8 z8 = {0, 0, 0, 0, 0, 0, 0, 0};
    // 6-arg form (amdgpu-toolchain clang-23): (g0, g1, g2, g3, g_extra, cpol)
    __builtin_amdgcn_tensor_load_to_lds(g0, g1, z4, z4, z8, 0);
}

__global__ void __launch_bounds__(BLOCK)
topo_hist_kernel(const int* __restrict__ prev, const int* __restrict__ curr,
                 float* __restrict__ out, unsigned* __restrict__ partials, int n) {
    __shared__ unsigned hist[BINS];        // 16 KB privatized histogram
    __shared__ int pbuf[2][CHUNK];         // 16 KB double buffer (prev)
    __shared__ int cbuf[2][CHUNK];         // 16 KB double buffer (curr)

    const int tid = threadIdx.x;
    for (int i = tid; i < BINS; i += BLOCK) hist[i] = 0u;

    const long long evPerBlock = ((long long)n + gridDim.x - 1) / gridDim.x;
    const long long start = (long long)blockIdx.x * evPerBlock;
    long long end = start + evPerBlock;
    if (end > (long long)n) end = (long long)n;
    const long long count = (end > start) ? (end - start) : 0;
    const int nchunks = (int)((count + CHUNK - 1) / CHUNK);

    const bool issuer = (tid < warpSize);  // wave 0 drives the TDM

    if (issuer && nchunks > 0) {
        long long c0 = count < (long long)CHUNK ? count : (long long)CHUNK;
        tdm_load_1d(lds_byte_offset(&pbuf[0][0]), prev + start, (unsigned)c0);
        tdm_load_1d(lds_byte_offset(&cbuf[0][0]), curr + start, (unsigned)c0);
    }
    __syncthreads();  // also covers hist zero-init

    for (int k = 0; k < nchunks; ++k) {
        const long long cstart = start + (long long)k * CHUNK;
        long long rem = end - cstart;
        const int csize = rem < (long long)CHUNK ? (int)rem : CHUNK;

        if (issuer) {
            if (k + 1 < nchunks) {
                const long long ns = start + (long long)(k + 1) * CHUNK;
                long long nrem = end - ns;
                const unsigned nsz = nrem < (long long)CHUNK ? (unsigned)nrem
                                                             : (unsigned)CHUNK;
                const int nb = (k + 1) & 1;
                tdm_load_1d(lds_byte_offset(&pbuf[nb][0]), prev + ns, nsz);
                tdm_load_1d(lds_byte_offset(&cbuf[nb][0]), curr + ns, nsz);
                // pair k+1 in flight (2 ops); <=2 outstanding => pair k complete
                __builtin_amdgcn_s_wait_tensorcnt(2);
            } else {
                __builtin_amdgcn_s_wait_tensorcnt(0);
            }
        }
        __syncthreads();  // publish chunk k to all waves

        const int b = k & 1;
#pragma unroll
        for (int j = 0; j < CHUNK / BLOCK; ++j) {
            const int idx = tid + j * BLOCK;  // lane-contiguous: bank-conflict-free reads
            if (idx < csize) {
                const unsigned p = (unsigned)pbuf[b][idx];
                const unsigned c = (unsigned)cbuf[b][idx];
                atomicAdd(&hist[p * NUM_TILES + c], 1u);   // ds_add_u32 (no return)
            }
        }
        __syncthreads();  // buffer b is free for the TDM write at iteration k+1
    }

    if (partials != nullptr) {
        // Two-stage path: plain coalesced stores, zero global-atomic contention.
        unsigned* my = partials + (size_t)blockIdx.x * BINS;
        for (int i = tid; i < BINS; i += BLOCK) my[i] = hist[i];
    } else {
        // Fallback: direct float atomic merge (exact for integer-valued sums).
        for (int i = tid; i < BINS; i += BLOCK) {
            const unsigned v = hist[i];
            if (v) atomicAdd(&out[i], (float)v);           // global_atomic_add_f32
        }
    }
}

// Sum per-block u32 partials into the (already seeded) float output.
// gridDim = (BINS/BLOCK, SLICES); each block sums nblocks/SLICES partials
// for 256 consecutive bins -> 8 MB streamed read, only BINS*SLICES atomics.
__global__ void __launch_bounds__(BLOCK)
topo_reduce_kernel(const unsigned* __restrict__ partials,
                   float* __restrict__ out, int nblocks) {
    const int bin = blockIdx.x * BLOCK + threadIdx.x;
    const int per = (nblocks + gridDim.y - 1) / gridDim.y;
    const int b0 = blockIdx.y * per;
    int b1 = b0 + per;
    if (b1 > nblocks) b1 = nblocks;

    unsigned s = 0;
    for (int b = b0; b < b1; ++b) {
        s += partials[(size_t)b * BINS + bin];   // lane-consecutive => coalesced
    }
    if (s) atomicAdd(&out[bin], (float)s);       // 32K atomics total: negligible
}

__global__ void topo_init_kernel(const float* __restrict__ transitions,
                                 const float* __restrict__ total,
                                 float* __restrict__ out, int n_events) {
    const int i = blockIdx.x * blockDim.x + threadIdx.x;
    if (i < BINS) {
        out[i] = transitions[i];
    } else if (i == BINS) {
        out[BINS] = total[0] + (float)n_events;
    }
}

extern "C" void kernel_launch(void* const* d_in, const int* in_sizes, int n_in,
                              void* d_out, int out_size, void* d_ws, size_t ws_size,
                              hipStream_t stream) {
    (void)n_in; (void)out_size;
    const int*   prev        = (const int*)d_in[0];
    const int*   curr        = (const int*)d_in[1];
    const float* transitions = (const float*)d_in[2];
    const float* total       = (const float*)d_in[3];
    float*       out         = (float*)d_out;
    const int n = in_sizes[0];

    // Two-stage merge needs NBLOCKS private histograms in workspace (8 MB).
    const size_t need = (size_t)NBLOCKS * BINS * sizeof(unsigned);
    unsigned* partials = (d_ws != nullptr && ws_size >= need) ? (unsigned*)d_ws
                                                              : nullptr;

    // Seed output with input transitions + total (runs first on the stream).
    topo_init_kernel<<<(BINS + 1 + BLOCK - 1) / BLOCK, BLOCK, 0, stream>>>(
        transitions, total, out, n);

    // TDM-streamed privatized histogram.
    topo_hist_kernel<<<NBLOCKS, BLOCK, 0, stream>>>(prev, curr, out, partials, n);

    if (partials != nullptr) {
        topo_reduce_kernel<<<dim3(BINS / BLOCK, SLICES), BLOCK, 0, stream>>>(
            partials, out, NBLOCKS);
    }
}